// TemporalAlignment_48902497632797
// MI455X (gfx1250) — compile-verified
//
#include <hip/hip_runtime.h>
#include <hip/hip_bf16.h>

#define BATCH 4
#define TP    2048
#define TE    16384
#define DIM   64

// ---- CDNA5 async global->LDS path (guarded so it compiles on any toolchain) ----
#if defined(__gfx1250__)
#if __has_builtin(__builtin_amdgcn_global_load_async_to_lds_b128)
#define TA_ASYNC_B128 1
#endif
#if __has_builtin(__builtin_amdgcn_global_load_async_to_lds_b32)
#define TA_ASYNC_B32 1
#endif
#endif

#if defined(TA_ASYNC_B128) || defined(TA_ASYNC_B32)
typedef int ta_v4i __attribute__((ext_vector_type(4)));
typedef __attribute__((address_space(1))) ta_v4i* ta_g128;  // global int4*
typedef __attribute__((address_space(3))) ta_v4i* ta_l128;  // LDS int4*
typedef __attribute__((address_space(1))) int*    ta_g32;   // global int*
typedef __attribute__((address_space(3))) int*    ta_l32;   // LDS int*

__device__ __forceinline__ void ta_wait_async() {
#if __has_builtin(__builtin_amdgcn_s_wait_asynccnt)
  __builtin_amdgcn_s_wait_asynccnt(0);
#else
  asm volatile("s_wait_asynccnt 0x0" ::: "memory");
#endif
}
#endif

// ------------------------------------------------------------------
// Kernel 0: zero the sums + counts region of workspace
// ------------------------------------------------------------------
__global__ void ta_zero(float* __restrict__ ws, int n) {
  int i = blockIdx.x * blockDim.x + threadIdx.x;
  int stride = gridDim.x * blockDim.x;
  for (; i < n; i += stride) ws[i] = 0.0f;
}

// ------------------------------------------------------------------
// Kernel 1: nearest price bar per event (brute-force argmin over LDS)
// grid = BATCH * (TE/256), block = 256 (8 waves of 32)
// ------------------------------------------------------------------
__global__ __launch_bounds__(256)
void ta_nearest(const float* __restrict__ price_ts,
                const float* __restrict__ event_ts,
                int*   __restrict__ idx_out,
                float* __restrict__ counts) {
  __shared__ __align__(16) float s_price[TP];

  const int tid   = threadIdx.x;
  const int b     = blockIdx.x / (TE / 256);
  const int chunk = blockIdx.x % (TE / 256);
  const float* gp = price_ts + b * TP;

  // Stage the batch's 2048 price timestamps (8 KB) into LDS.
#if defined(TA_ASYNC_B128)
  for (int i = tid; i < TP / 4; i += 256) {
    __builtin_amdgcn_global_load_async_to_lds_b128(
        (ta_g128)(gp + 4 * i), (ta_l128)(&s_price[4 * i]), 0, 0);
  }
  ta_wait_async();
#elif defined(TA_ASYNC_B32)
  for (int i = tid; i < TP; i += 256) {
    __builtin_amdgcn_global_load_async_to_lds_b32(
        (ta_g32)(gp + i), (ta_l32)(&s_price[i]), 0, 0);
  }
  ta_wait_async();
#else
  for (int i = tid; i < TP; i += 256) s_price[i] = gp[i];
#endif
  __syncthreads();

  const int   e  = chunk * 256 + tid;
  const float te = event_ts[b * TE + e];

  float best = 3.4e38f;
  int   bi   = 0;
  const float4* sp4 = (const float4*)s_price;
#pragma unroll 4
  for (int i = 0; i < TP / 4; ++i) {
    float4 p = sp4[i];                 // uniform address -> LDS broadcast
    float d0 = fabsf(te - p.x);
    float d1 = fabsf(te - p.y);
    float d2 = fabsf(te - p.z);
    float d3 = fabsf(te - p.w);
    if (d0 < best) { best = d0; bi = 4 * i + 0; }
    if (d1 < best) { best = d1; bi = 4 * i + 1; }
    if (d2 < best) { best = d2; bi = 4 * i + 2; }
    if (d3 < best) { best = d3; bi = 4 * i + 3; }
  }

  idx_out[b * TE + e] = bi;
  atomicAdd(&counts[b * TP + bi], 1.0f);
}

// ------------------------------------------------------------------
// Kernel 2: scatter-add event values into per-bar sums
// one thread per (b, e, d); d fastest -> coalesced
// ------------------------------------------------------------------
__global__ __launch_bounds__(256)
void ta_scatter(const float* __restrict__ ev,
                const int*   __restrict__ idx,
                float*       __restrict__ sums) {
  int g  = blockIdx.x * 256 + threadIdx.x;   // (b*TE + e)*64 + d
  int d  = g & (DIM - 1);
  int be = g >> 6;                           // b*TE + e
  int b  = be >> 14;                         // TE = 2^14
  int t  = idx[be];
  atomicAdd(&sums[(((b * TP) + t) << 6) + d], ev[g]);
}

// ------------------------------------------------------------------
// Kernel 3: normalize and emit aligned + coverage
// ------------------------------------------------------------------
__global__ __launch_bounds__(256)
void ta_finalize(const float* __restrict__ sums,
                 const float* __restrict__ counts,
                 float* __restrict__ aligned,
                 float* __restrict__ cover) {
  int g   = blockIdx.x * 256 + threadIdx.x;  // row*64 + d
  int d   = g & (DIM - 1);
  int row = g >> 6;                          // b*TP + t
  float c   = counts[row];
  bool  has = c > 0.0f;
  aligned[g] = has ? sums[g] / c : 0.0f;
  if (d == 0) cover[row] = has ? 1.0f : 0.0f;
}

// ------------------------------------------------------------------
extern "C" void kernel_launch(void* const* d_in, const int* in_sizes, int n_in,
                              void* d_out, int out_size, void* d_ws, size_t ws_size,
                              hipStream_t stream) {
  (void)in_sizes; (void)n_in; (void)out_size; (void)ws_size;

  const float* price = (const float*)d_in[0];  // (B, TP)
  const float* evts  = (const float*)d_in[1];  // (B, TE)
  const float* evv   = (const float*)d_in[2];  // (B, TE, D)

  float* out_aligned = (float*)d_out;                       // B*TP*DIM floats
  float* out_cover   = out_aligned + BATCH * TP * DIM;      // B*TP floats

  float* sums   = (float*)d_ws;                             // B*TP*DIM
  float* counts = sums + BATCH * TP * DIM;                  // B*TP
  int*   idx    = (int*)(counts + BATCH * TP);              // B*TE

  const int nzero = BATCH * TP * DIM + BATCH * TP;
  ta_zero<<<256, 256, 0, stream>>>(sums, nzero);
  ta_nearest<<<BATCH * (TE / 256), 256, 0, stream>>>(price, evts, idx, counts);
  ta_scatter<<<(BATCH * TE * DIM) / 256, 256, 0, stream>>>(evv, idx, sums);
  ta_finalize<<<(BATCH * TP * DIM) / 256, 256, 0, stream>>>(sums, counts,
                                                            out_aligned, out_cover);
}